// LoNet_26723286516258
// MI455X (gfx1250) — compile-verified
//
#include <hip/hip_runtime.h>

#define NM 8                       // modes
#define NP 7                       // photons
#define NB 8                       // basic states (batch)
#define MM (1L << 21)              // 8^7 packed Fock indices
#define BM ((long)NB * MM)         // 16,777,216 complex elements total

typedef float v2f __attribute__((ext_vector_type(2)));
typedef float v8f __attribute__((ext_vector_type(8)));

__constant__ float kFact[8] = {1.f, 1.f, 2.f, 6.f, 24.f, 120.f, 720.f, 5040.f};

// base-8 repunit with c ones, branch/table-free
__device__ inline unsigned rep8(unsigned c) {
  return 0x249249u & ((1u << (3u * c)) - 1u);
}

// Full build (used only in the small table kernel): sorted pack + sqrt(prod n_k!)
__device__ inline void target_scale_full(unsigned cnt, long& target, float& scale) {
  long t = 0; float p = 1.f;
#pragma unroll
  for (int k = 0; k < NM; ++k) {
    unsigned ck = (cnt >> (4 * k)) & 15u;
    t = (t << (3 * ck)) + (long)((unsigned)k * rep8(ck));
    p *= kFact[ck];
  }
  target = t; scale = sqrtf(p);
}

// ---------------------------------------------------------------------------
// Kernel 1: build the 8x8 complex unitary (Clements decomposition + phases)
// ---------------------------------------------------------------------------
__global__ void k_build_unitary(const float* __restrict__ alphas,
                                const float* __restrict__ betas,
                                const float* __restrict__ gammas,
                                float2* __restrict__ Uc) {   // [8][8] complex
  if (threadIdx.x != 0 || blockIdx.x != 0) return;
  float Ur[NM][NM], Ui[NM][NM];
  for (int r = 0; r < NM; ++r)
    for (int c = 0; c < NM; ++c) { Ur[r][c] = (r == c) ? 1.f : 0.f; Ui[r][c] = 0.f; }

  for (int i = 0; i < NM - 1; ++i) {
    for (int j = i + 1; j < NM; ++j) {
      int k = i * (2 * NM - 3 - i) / 2 + j - 1;
      float sa = sinf(alphas[k]), ca = cosf(alphas[k]);
      float cb = cosf(betas[k]),  sb = sinf(betas[k]);
      float lijr =  cb * sa, liji = -sb * sa;   // conj(e^{ib}) sin
      float ljir = -cb * sa, ljii = -sb * sa;   // -e^{ib} sin
      for (int c = 0; c < NM; ++c) {
        float air = Ur[i][c], aii = Ui[i][c];
        float ajr = Ur[j][c], aji = Ui[j][c];
        Ur[i][c] = ca * air + lijr * ajr - liji * aji;
        Ui[i][c] = ca * aii + lijr * aji + liji * ajr;
        Ur[j][c] = ljir * air - ljii * aii + ca * ajr;
        Ui[j][c] = ljir * aii + ljii * air + ca * aji;
      }
    }
  }
  for (int r = 0; r < NM; ++r) {
    float gr = 1.f, gi = 0.f;
    if (r < NM - 1) { gr = cosf(gammas[r]); gi = sinf(gammas[r]); }
    for (int c = 0; c < NM; ++c) {
      float ur = Ur[r][c], ui = Ui[r][c];
      float2 z; z.x = gr * ur - gi * ui; z.y = gr * ui + gi * ur;
      Uc[r * NM + c] = z;
    }
  }
}

// ---------------------------------------------------------------------------
// Kernel 2: per-batch rank-1 factor tables + lo digit-count table
// ---------------------------------------------------------------------------
__global__ void k_tables(const int* __restrict__ states,
                         const float2* __restrict__ Uc,
                         float2* __restrict__ uTab,      // [NB][512]
                         float2* __restrict__ vTab,      // [NB][4096]
                         unsigned* __restrict__ cntLo) { // [4096]
  const int b = blockIdx.x;
  const int t = threadIdx.x;
  __shared__ float2 U[64];
  __shared__ int d[NP];
  if (t < 64) U[t] = Uc[t];
  if (t < NP) d[t] = states[b * NP + t];
  __syncthreads();

  unsigned cb = 0;
  for (int j = 0; j < NP; ++j) cb += 1u << (4 * d[j]);
  long dummy; float nrm;
  target_scale_full(cb, dummy, nrm);     // sqrt(prod n_k!)
  nrm = 1.f / nrm;                       // norm_b

  for (int h = t; h < 512; h += 256) {
    int h1 = h >> 6, h2 = (h >> 3) & 7, h3 = h & 7;
    float2 a = U[h1 * 8 + d[0]], bb = U[h2 * 8 + d[1]], c = U[h3 * 8 + d[2]];
    float2 p; p.x = a.x * bb.x - a.y * bb.y; p.y = a.x * bb.y + a.y * bb.x;
    float2 q; q.x = p.x * c.x - p.y * c.y;   q.y = p.x * c.y + p.y * c.x;
    q.x *= nrm; q.y *= nrm;
    uTab[b * 512 + h] = q;
  }
  for (int l = t; l < 4096; l += 256) {
    int l1 = l >> 9, l2 = (l >> 6) & 7, l3 = (l >> 3) & 7, l4 = l & 7;
    float2 a = U[l1 * 8 + d[3]], bb = U[l2 * 8 + d[4]];
    float2 c = U[l3 * 8 + d[5]], e  = U[l4 * 8 + d[6]];
    float2 p; p.x = a.x * bb.x - a.y * bb.y; p.y = a.x * bb.y + a.y * bb.x;
    float2 q; q.x = p.x * c.x - p.y * c.y;   q.y = p.x * c.y + p.y * c.x;
    float2 r; r.x = q.x * e.x - q.y * e.y;   r.y = q.x * e.y + q.y * e.x;
    vTab[b * 4096 + l] = r;
    if (b == 0)
      cntLo[l] = (1u << (4 * l1)) + (1u << (4 * l2)) + (1u << (4 * l3)) + (1u << (4 * l4));
  }
}

// ---------------------------------------------------------------------------
// Zero a buffer with float4 stores
// ---------------------------------------------------------------------------
__global__ void k_zero4(float4* __restrict__ p, long n4) {
  long i = (long)blockIdx.x * blockDim.x + threadIdx.x;
  if (i < n4) p[i] = make_float4(0.f, 0.f, 0.f, 0.f);
}

// ---------------------------------------------------------------------------
// Fused rank-1 outer product (WMMA) + symmetrizing atomic scatter.
// One wave = 16 hi x 64 lo-complex = 8 WMMAs, A held across all 8.
// Scatter epilogue: per sub-tile build the 6-digit sorted pack once, then
// insert the last hi digit (== v) per element via nibble prefix sums.
// ---------------------------------------------------------------------------
__global__ void __launch_bounds__(256)
k_outer(const float2* __restrict__ uTab, const float2* __restrict__ vTab,
        const unsigned* __restrict__ cntLo, float* __restrict__ out) {
  __shared__ float sFact[8];   // c!
  __shared__ float sSq[8];     // sqrt(c+1)
  if (threadIdx.x < 8) {
    const float f[8] = {1.f, 1.f, 2.f, 6.f, 24.f, 120.f, 720.f, 5040.f};
    sFact[threadIdx.x] = f[threadIdx.x];
    sSq[threadIdx.x]   = sqrtf((float)(threadIdx.x + 1));
  }
  __syncthreads();

  const int  lane   = threadIdx.x & 31;
  const long wv     = ((long)blockIdx.x * 8) + (threadIdx.x >> 5);
  const int  b      = (int)(wv >> 11);          // 8 batches
  const int  rem    = (int)(wv & 2047);
  const int  hi0    = (rem >> 6) << 4;          // 32 hi-tiles of 16
  const int  loBase = (rem & 63) << 6;          // 64 lo-supertiles of 64 complex

  const int  nIdx   = lane & 15;
  const bool hiHalf = lane >= 16;               // K half {2,3}: zero operands

  // A operand (held across all 8 WMMAs): rows m = nIdx, K0=u_re, K1=u_im
  float2 uu = uTab[b * 512 + hi0 + nIdx];
  v2f a;
  a.x = hiHalf ? 0.f : uu.x;
  a.y = hiHalf ? 0.f : uu.y;

  const int c   = nIdx >> 1;                    // lo offset within sub-tile
  const int par = nIdx & 1;                     // 0 = Re, 1 = Im
  const int mb  = hiHalf ? 8 : 0;

  // first two hi digits are fixed per lane (hi0+mb is 8-aligned, last digit = v)
  const int hib = hi0 + mb;
  const unsigned cnt2 = (1u << (4 * (hib >> 6))) + (1u << (4 * ((hib >> 3) & 7)));
  const long obase = ((long)b << 21);

#pragma unroll
  for (int sub = 0; sub < 8; ++sub) {
    const int lo0 = loBase + (sub << 3);
    float2 vv = vTab[b * 4096 + lo0 + c];
    float b0 = par ? vv.y : vv.x;
    float b1 = par ? vv.x : -vv.y;
    v2f bb;
    bb.x = hiHalf ? 0.f : b0;
    bb.y = hiHalf ? 0.f : b1;

    v8f acc = {0.f, 0.f, 0.f, 0.f, 0.f, 0.f, 0.f, 0.f};
    acc = __builtin_amdgcn_wmma_f32_16x16x4_f32(false, a, false, bb,
                                                (short)0, acc, false, false);

    // ---- 6-digit base (2 hi digits + 4 lo digits), built once per sub ----
    const unsigned cbase = cnt2 + cntLo[lo0 + c];
    unsigned T6 = 0; float p6 = 1.f;
#pragma unroll
    for (int k = 0; k < NM; ++k) {
      unsigned ck = (cbase >> (4 * k)) & 15u;
      T6 = (T6 << (3 * ck)) + (unsigned)k * rep8(ck);
      p6 *= sFact[ck];
    }
    const float S6 = sqrtf(p6);
    const unsigned ps = cbase * 0x11111110u;    // nibble prefix sums (digits < v)

#pragma unroll
    for (int v = 0; v < 8; ++v) {
      const unsigned nlt = (ps >> (4 * v)) & 15u;
      const unsigned sh  = 18u - 3u * nlt;      // 3*(6 - nlt)
      const unsigned low  = T6 & ((1u << sh) - 1u);
      const unsigned high = T6 >> sh;
      const unsigned target = (((high << 3) + (unsigned)v) << sh) + low;
      const unsigned cv = (cbase >> (4 * v)) & 15u;
      const float scl = S6 * sSq[cv];           // S6 * sqrt(cv+1)
      const long o = 2 * (obase + (long)target) + par;
      unsafeAtomicAdd(out + o, acc[v] * scl);   // global_atomic_add_f32
    }
  }
}

// ---------------------------------------------------------------------------
extern "C" void kernel_launch(void* const* d_in, const int* in_sizes, int n_in,
                              void* d_out, int out_size, void* d_ws, size_t ws_size,
                              hipStream_t stream) {
  const float* alphas = (const float*)d_in[0];   // 28
  const float* betas  = (const float*)d_in[1];   // 28
  const float* gammas = (const float*)d_in[2];   // 8
  const int*   states = (const int*)d_in[3];     // 8x7

  float2*   Uc    = (float2*)d_ws;                            // 512 B
  unsigned* cntLo = (unsigned*)((char*)d_ws + 4096);          // 16 KB
  float2*   uTab  = (float2*)((char*)d_ws + 32768);           // 32 KB
  float2*   vTab  = (float2*)((char*)d_ws + 65536);           // 256 KB

  k_build_unitary<<<1, 32, 0, stream>>>(alphas, betas, gammas, Uc);
  k_tables<<<NB, 256, 0, stream>>>(states, Uc, uTab, vTab, cntLo);

  const long n4 = BM * 2 / 4;                                 // 8,388,608 float4
  k_zero4<<<(int)((n4 + 255) / 256), 256, 0, stream>>>((float4*)d_out, n4);

  // 8 batches x 32 hi-tiles x 64 lo-supertiles = 16384 waves, 8 waves/block
  k_outer<<<2048, 256, 0, stream>>>(uTab, vTab, cntLo, (float*)d_out);
}